// Normal_VM_NoLayer_11141145166391
// MI455X (gfx1250) — compile-verified
//
#include <hip/hip_runtime.h>
#include <math.h>

typedef __attribute__((ext_vector_type(2))) float v2f;
typedef __attribute__((ext_vector_type(8))) float v8f;

#define KAPPA      2.0f
#define S_TOT      64     // s_in * nh
#define NV         4
#define NC         8
#define PD         16     // N_PHI * N_DIST
#define PI_F       3.14159265358979323846f

// One wave (32 lanes) per (b,n) site.
// Per (site, v) compute D[16pd x 16] = W'^T[16x64] * B[64x16] with
// V_WMMA_F32_16X16X4_F32 chains, where B cols 0..7 = x, col 8 = ones so
// D[:,8] is the masked-normalization denominator (exact, incl. 1e-10 floor).
// Loop order: k outer, v inner with 4 independent accumulators so the
// v-invariant W fragment is loaded once per k-chunk and the 4 WMMAs pipeline.
__launch_bounds__(32)
__global__ void vm_normal_wmma_kernel(const float* __restrict__ dists,
                                      const float* __restrict__ angles,
                                      const float* __restrict__ x,
                                      const int*   __restrict__ mask,
                                      const float* __restrict__ sigma_p,
                                      float* __restrict__ out,
                                      int nsites)
{
    __shared__ float ws_t[PD * S_TOT];      // w[pd][s]   K-contiguous  (4 KB)
    __shared__ float kp_t[NV * S_TOT];      // keep[v][s] K-contiguous  (1 KB)
    __shared__ float xs[S_TOT * NV * NC];   // x[s][v][c]               (8 KB)

    const int site = blockIdx.x;
    const int lane = threadIdx.x;           // wave32

    const float sig    = fmaxf(sigma_p[0], 1e-10f);
    const float invsig = 1.0f / sig;

    const float phis[4] = { -PI_F, -0.5f * PI_F, 0.0f, 0.5f * PI_F };
    const float mus[4]  = { 0.0f, 0.1f / 3.0f, 0.2f / 3.0f, 0.1f };

    // ---- weights: 2 sources per lane, fast transcendentals ---------------
    #pragma unroll
    for (int t = 0; t < 2; ++t) {
        const int   s  = lane + 32 * t;
        const float dd = dists[site * S_TOT + s];
        const float aa = angles[site * S_TOT + s];
        float wvm[4], wnd[4];
        #pragma unroll
        for (int p = 0; p < 4; ++p) {
            const float c = __cosf(aa - phis[p]);        // v_cos_f32
            const float e = __expf(KAPPA * (c - 1.0f));  // v_exp_f32
            wvm[p] = (dd == 0.0f) ? 1.0f : e;            // dist==0 -> w_vm = 1
        }
        #pragma unroll
        for (int d = 0; d < 4; ++d) {
            const float z = (dd - mus[d]) * invsig;
            wnd[d] = __expf(-0.5f * z * z);
        }
        #pragma unroll
        for (int p = 0; p < 4; ++p)
            #pragma unroll
            for (int d = 0; d < 4; ++d)
                ws_t[(p * 4 + d) * S_TOT + s] = wvm[p] * wnd[d];
    }

    // ---- keep mask, transposed: 8 coalesced loads per lane ---------------
    #pragma unroll
    for (int i = 0; i < 8; ++i) {
        const int idx = lane * 8 + i;                    // idx = s*4 + v
        const int s   = idx >> 2;
        const int v   = idx & 3;
        kp_t[v * S_TOT + s] =
            (mask[site * (S_TOT * NV) + idx] != 0) ? 0.0f : 1.0f;
    }

    // ---- stage x (2048 floats) via coalesced 128-bit moves ---------------
    {
        const float4* xg = (const float4*)(x + (size_t)site * (S_TOT * NV * NC));
        float4*       xl = (float4*)xs;
        #pragma unroll
        for (int i = 0; i < 16; ++i)
            xl[lane + i * 32] = xg[lane + i * 32];
    }

    __syncthreads();

    // ---- WMMA fragment indexing (32-bit A 16x4 / B 4x16 layouts) ---------
    // lane L: row/col = L%16 ; holds K = 2*(L/16) + {0,1} in the two dwords.
    const int   m     = lane & 15;      // A: M (=pd row); B/D: N (=c col)
    const int   khalf = lane >> 4;      // 0 or 1
    const float selX  = (m < 8)  ? 1.0f : 0.0f;   // B data columns
    const float selO  = (m == 8) ? 1.0f : 0.0f;   // B ones column (denominator)
    const int   cidx  = m & 7;
    const int   srcln = (lane & 16) | 8;          // lane holding D[:,8]

    v8f acc0 = {0.f,0.f,0.f,0.f,0.f,0.f,0.f,0.f};
    v8f acc1 = acc0, acc2 = acc0, acc3 = acc0;

    const int kofs = khalf * 2;
    for (int k0 = 0; k0 < S_TOT; k0 += 4) {
        const int k = k0 + kofs;
        // v-invariant W pair: one aligned ds_load_b64 (k is even)
        const v2f wv = *(const v2f*)&ws_t[m * S_TOT + k];

#define VM_STEP(V, ACC)                                                     \
        {                                                                   \
            const v2f kp = *(const v2f*)&kp_t[(V) * S_TOT + k];             \
            v2f a, b;                                                       \
            a.x = wv.x * kp.x + 1e-10f;                                     \
            a.y = wv.y * kp.y + 1e-10f;                                     \
            b.x = selX * xs[k * (NV * NC) + (V) * NC + cidx] + selO;        \
            b.y = selX * xs[(k + 1) * (NV * NC) + (V) * NC + cidx] + selO;  \
            ACC = __builtin_amdgcn_wmma_f32_16x16x4_f32(                    \
                false, a, false, b, (short)0, ACC, false, false);           \
        }

        VM_STEP(0, acc0)
        VM_STEP(1, acc1)
        VM_STEP(2, acc2)
        VM_STEP(3, acc3)
#undef VM_STEP
    }

    // ---- normalize and store ---------------------------------------------
    // D layout: vgpr j -> M = j + 8*khalf, N = lane&15; denominator at N==8.
#define VM_STORE(V, ACC)                                                    \
    {                                                                       \
        _Pragma("unroll")                                                   \
        for (int j = 0; j < 8; ++j) {                                       \
            const float num = ACC[j];                                       \
            const float den = __shfl(num, srcln, 32);                       \
            if (m < 8) {                                                    \
                const int mrow = j + khalf * 8;   /* pd = p*4+d */          \
                out[(size_t)site * 512 + (V) * 128 + mrow * 8 + m] =        \
                    num / den;                                              \
            }                                                               \
        }                                                                   \
    }

    VM_STORE(0, acc0)
    VM_STORE(1, acc1)
    VM_STORE(2, acc2)
    VM_STORE(3, acc3)
#undef VM_STORE

    // ---- m2: cell fully masked iff all 64 sources masked -----------------
    if (lane < NV) {
        float ssum = 0.0f;
        for (int s = 0; s < S_TOT; ++s)
            ssum += kp_t[lane * S_TOT + s];
        out[(size_t)nsites * 512 + (size_t)site * NV + lane] =
            (ssum == 0.0f) ? 1.0f : 0.0f;
    }
}

extern "C" void kernel_launch(void* const* d_in, const int* in_sizes, int n_in,
                              void* d_out, int out_size, void* d_ws, size_t ws_size,
                              hipStream_t stream) {
    const float* dists  = (const float*)d_in[0];   // (b,n,r,s_in,nh)
    const float* angles = (const float*)d_in[1];   // (b,n,r,s_in,nh)
    const float* x      = (const float*)d_in[2];   // (b,n,r,S,nv,nc)
    const int*   mask   = (const int*)d_in[3];     // (b,n,r,S,nv) bool->int
    const float* sigma  = (const float*)d_in[4];   // scalar

    const int nsites = in_sizes[0] / S_TOT;        // b*n*r

    vm_normal_wmma_kernel<<<nsites, 32, 0, stream>>>(
        dists, angles, x, mask, sigma, (float*)d_out, nsites);
}